// TriangleMultiplication_88115549045151
// MI455X (gfx1250) — compile-verified
//
#include <hip/hip_runtime.h>
#include <math.h>

// ---------------------------------------------------------------------------
// Triangle multiplication (outgoing), fp32, exact-precision WMMA path.
//   B=4, N=256, D_Z=D_H=128.
// Workspace use: left[B,128,N,N] + right[B,128,N,N] fp32 = 268 MB.
// d_out ([B,N,N,128]) is used as the triangle-matmul scratch, then rewritten
// in-place by the final fused layernorm/projection/gate kernel.
// All matmuls use V_WMMA_F32_16X16X4_F32 (exact fp32 on the matrix pipe).
// ---------------------------------------------------------------------------

typedef __attribute__((ext_vector_type(2))) float v2f;
typedef __attribute__((ext_vector_type(8))) float v8f;

#define WMMA_F32(a, b, c) \
    __builtin_amdgcn_wmma_f32_16x16x4_f32(false, (a), false, (b), (short)0, (c), false, false)

constexpr int kN  = 256;   // pair grid edge
constexpr int kD  = 128;   // d_z == d_h
constexpr int kLDP = 132;  // padded LDS row stride (floats) for 16x128 tiles
constexpr float kEps = 1e-5f;

__device__ __forceinline__ float sigmoidf_(float x) { return 1.0f / (1.0f + __expf(-x)); }

// ---------------------------------------------------------------------------
// Kernel 1: per-position layernorm + {lp,lg,rp,rg} projections + mask.
// Block = 256 thr (8 waves) handles 16 positions: (b, row, col0..col0+15).
// Wave w owns output-channel tile d in [16w, 16w+16); does all 4 projections.
// Writes left/right transposed to [b][d][prow][pcol] for the einsum.
// ---------------------------------------------------------------------------
__global__ __launch_bounds__(256) void k_norm_proj(
    const float* __restrict__ z, const unsigned char* __restrict__ mask,
    const float* __restrict__ nrmw, const float* __restrict__ nrmb,
    const float* __restrict__ lpw, const float* __restrict__ lpb,
    const float* __restrict__ lgw, const float* __restrict__ lgb,
    const float* __restrict__ rpw, const float* __restrict__ rpb,
    const float* __restrict__ rgw, const float* __restrict__ rgb,
    float* __restrict__ Lw, float* __restrict__ Rw)
{
    __shared__ float zin[16 * kLDP];

    const int tid  = threadIdx.x;
    const int lane = tid & 31;
    const int wave = tid >> 5;
    const int h    = lane >> 4;   // lane half (0/1)
    const int lm   = lane & 15;

    const int tileId = blockIdx.x;
    const int b    = tileId >> 12;        // (N*N/16) = 4096 tiles per batch
    const int rem  = tileId & 4095;
    const int row  = rem >> 4;
    const int col0 = (rem & 15) << 4;

    // ---- layernorm: wave handles positions 2*wave, 2*wave+1 ----
    #pragma unroll
    for (int rr = 0; rr < 2; ++rr) {
        const int m = wave * 2 + rr;
        const size_t zb = ((size_t)(b * kN + row) * kN + (col0 + m)) * kD;
        float v[4];
        float s = 0.f, s2 = 0.f;
        #pragma unroll
        for (int q = 0; q < 4; ++q) {
            v[q] = z[zb + lane + 32 * q];
            s += v[q]; s2 += v[q] * v[q];
        }
        #pragma unroll
        for (int off = 16; off > 0; off >>= 1) {
            s  += __shfl_xor(s,  off, 32);
            s2 += __shfl_xor(s2, off, 32);
        }
        const float mu  = s * (1.0f / kD);
        const float var = s2 * (1.0f / kD) - mu * mu;
        const float ri  = rsqrtf(var + kEps);
        #pragma unroll
        for (int q = 0; q < 4; ++q) {
            const int k = lane + 32 * q;
            zin[m * kLDP + k] = (v[q] - mu) * ri * nrmw[k] + nrmb[k];
        }
    }
    __syncthreads();

    // ---- 4 projections, K = 128 as 32 steps of K=4 fp32 WMMA ----
    const int d = wave * 16 + lm;   // output channel for this lane
    v8f alp = {}, alg = {}, arp = {}, arg_ = {};
    #pragma unroll 8
    for (int k0 = 0; k0 < kD; k0 += 4) {
        const int k = k0 + 2 * h;
        const v2f a   = *reinterpret_cast<const v2f*>(&zin[lm * kLDP + k]);
        const v2f wlp = *reinterpret_cast<const v2f*>(&lpw[d * kD + k]);
        const v2f wlg = *reinterpret_cast<const v2f*>(&lgw[d * kD + k]);
        const v2f wrp = *reinterpret_cast<const v2f*>(&rpw[d * kD + k]);
        const v2f wrg = *reinterpret_cast<const v2f*>(&rgw[d * kD + k]);
        alp  = WMMA_F32(a, wlp, alp);
        alg  = WMMA_F32(a, wlg, alg);
        arp  = WMMA_F32(a, wrp, arp);
        arg_ = WMMA_F32(a, wrg, arg_);
    }

    const float blp = lpb[d], blg = lgb[d], brp = rpb[d], brg = rgb[d];
    const float mrow = mask[b * kN + row] ? 1.f : 0.f;
    float* Lbase = Lw + ((size_t)(b * kD + d) * kN + row) * kN;
    float* Rbase = Rw + ((size_t)(b * kD + d) * kN + row) * kN;

    #pragma unroll
    for (int v = 0; v < 8; ++v) {
        const int m   = v + 8 * h;        // C/D layout: lanes 16-31 hold M+8
        const int col = col0 + m;
        const float pm = mrow * (mask[b * kN + col] ? 1.f : 0.f);
        Lbase[col] = (alp[v] + blp) * sigmoidf_(alg[v]  + blg) * pm;
        Rbase[col] = (arp[v] + brp) * sigmoidf_(arg_[v] + brg) * pm;
    }
}

// ---------------------------------------------------------------------------
// Kernel 2: 512 independent 256x256x256 fp32 matmuls (one per (b,d) slice).
// Block = 256 thr computes a 64x64 C tile; wave w -> 16x32 (two 16x16 tiles).
// K staged through LDS in 32-wide panels.  Output scattered back to
// [B,N,N,D] layout (scattered 4-byte stores; adjacent lanes are kD floats apart).
// ---------------------------------------------------------------------------
constexpr int kLT = 36;  // Lt row stride (floats)
constexpr int kRT = 68;  // Rt row stride (floats)

__global__ __launch_bounds__(256) void k_tri(
    const float* __restrict__ Lw, const float* __restrict__ Rw,
    float* __restrict__ out)
{
    __shared__ float Lt[64 * kLT];
    __shared__ float Rt[32 * kRT];

    const int tid  = threadIdx.x;
    const int lane = tid & 31;
    const int wave = tid >> 5;
    const int h    = lane >> 4;
    const int lm   = lane & 15;

    const int slice = blockIdx.z;               // b*128 + d
    const size_t sb = (size_t)slice * (kN * kN);
    const int row0 = blockIdx.y * 64;
    const int col0 = blockIdx.x * 64;

    const int mrow = (wave & 3) * 16;
    const int nc0  = (wave >> 2) * 32;

    v8f acc0 = {}, acc1 = {};

    for (int kc = 0; kc < kN; kc += 32) {
        __syncthreads();
        {   // Lt: 64 rows x 32 floats
            const int r  = tid >> 3;
            const int cq = (tid & 7) << 2;
            #pragma unroll
            for (int p = 0; p < 2; ++p) {
                const int rr = r + p * 32;
                const float4 v =
                    *reinterpret_cast<const float4*>(&Lw[sb + (size_t)(row0 + rr) * kN + kc + cq]);
                *reinterpret_cast<float4*>(&Lt[rr * kLT + cq]) = v;
            }
        }
        {   // Rt: 32 rows x 64 floats
            const int r  = tid >> 4;
            const int cq = (tid & 15) << 2;
            #pragma unroll
            for (int p = 0; p < 2; ++p) {
                const int rr = r + p * 16;
                const float4 v =
                    *reinterpret_cast<const float4*>(&Rw[sb + (size_t)(kc + rr) * kN + col0 + cq]);
                *reinterpret_cast<float4*>(&Rt[rr * kRT + cq]) = v;
            }
        }
        __syncthreads();

        #pragma unroll
        for (int kk = 0; kk < 32; kk += 4) {
            const int k = kk + 2 * h;
            const v2f a = *reinterpret_cast<const v2f*>(&Lt[(mrow + lm) * kLT + k]);
            v2f b0, b1;
            b0.x = Rt[k * kRT + nc0 + lm];
            b0.y = Rt[(k + 1) * kRT + nc0 + lm];
            b1.x = Rt[k * kRT + nc0 + 16 + lm];
            b1.y = Rt[(k + 1) * kRT + nc0 + 16 + lm];
            acc0 = WMMA_F32(a, b0, acc0);
            acc1 = WMMA_F32(a, b1, acc1);
        }
    }

    const int b = slice >> 7;
    const int d = slice & 127;
    #pragma unroll
    for (int v = 0; v < 8; ++v) {
        const int i  = row0 + mrow + v + 8 * h;
        const int j0 = col0 + nc0 + lm;
        out[((size_t)(b * kN + i) * kN + j0) * kD + d]        = acc0[v];
        out[((size_t)(b * kN + i) * kN + j0 + 16) * kD + d]   = acc1[v];
    }
}

// ---------------------------------------------------------------------------
// Kernel 3: layernorm(tri) @ op_w.T + op_b, gated by sigmoid(z @ og_w.T + og_b).
// Operates in-place on d_out (reads complete before the sync; writes after).
// ---------------------------------------------------------------------------
__global__ __launch_bounds__(256) void k_out(
    const float* __restrict__ z,
    const float* __restrict__ onw, const float* __restrict__ onb,
    const float* __restrict__ opw, const float* __restrict__ opb,
    const float* __restrict__ ogw, const float* __restrict__ ogb,
    float* __restrict__ out)
{
    __shared__ float tin[16 * kLDP];
    __shared__ float zr[16 * kLDP];

    const int tid  = threadIdx.x;
    const int lane = tid & 31;
    const int wave = tid >> 5;
    const int h    = lane >> 4;
    const int lm   = lane & 15;

    const int tileId = blockIdx.x;
    const int b    = tileId >> 12;
    const int rem  = tileId & 4095;
    const int row  = rem >> 4;
    const int col0 = (rem & 15) << 4;

    // raw z tile -> LDS (for the gate projection)
    {
        const int r = tid >> 4;
        const int c = (tid & 15) << 3;
        const size_t zb = ((size_t)(b * kN + row) * kN + (col0 + r)) * kD + c;
        const float4 v0 = *reinterpret_cast<const float4*>(&z[zb]);
        const float4 v1 = *reinterpret_cast<const float4*>(&z[zb + 4]);
        *reinterpret_cast<float4*>(&zr[r * kLDP + c])     = v0;
        *reinterpret_cast<float4*>(&zr[r * kLDP + c + 4]) = v1;
    }

    // layernorm of triangle result -> tin
    #pragma unroll
    for (int rr = 0; rr < 2; ++rr) {
        const int m = wave * 2 + rr;
        const size_t ob = ((size_t)(b * kN + row) * kN + (col0 + m)) * kD;
        float v[4];
        float s = 0.f, s2 = 0.f;
        #pragma unroll
        for (int q = 0; q < 4; ++q) {
            v[q] = out[ob + lane + 32 * q];
            s += v[q]; s2 += v[q] * v[q];
        }
        #pragma unroll
        for (int off = 16; off > 0; off >>= 1) {
            s  += __shfl_xor(s,  off, 32);
            s2 += __shfl_xor(s2, off, 32);
        }
        const float mu  = s * (1.0f / kD);
        const float var = s2 * (1.0f / kD) - mu * mu;
        const float ri  = rsqrtf(var + kEps);
        #pragma unroll
        for (int q = 0; q < 4; ++q) {
            const int k = lane + 32 * q;
            tin[m * kLDP + k] = (v[q] - mu) * ri * onw[k] + onb[k];
        }
    }
    __syncthreads();   // all in-place reads of `out` complete here

    const int d = wave * 16 + lm;
    v8f aop = {}, aog = {};
    #pragma unroll 8
    for (int k0 = 0; k0 < kD; k0 += 4) {
        const int k = k0 + 2 * h;
        const v2f at  = *reinterpret_cast<const v2f*>(&tin[lm * kLDP + k]);
        const v2f az  = *reinterpret_cast<const v2f*>(&zr[lm * kLDP + k]);
        const v2f wop = *reinterpret_cast<const v2f*>(&opw[d * kD + k]);
        const v2f wog = *reinterpret_cast<const v2f*>(&ogw[d * kD + k]);
        aop = WMMA_F32(at, wop, aop);
        aog = WMMA_F32(az, wog, aog);
    }

    const float bop = opb[d], bog = ogb[d];
    #pragma unroll
    for (int v = 0; v < 8; ++v) {
        const int col = col0 + v + 8 * h;
        out[((size_t)(b * kN + row) * kN + col) * kD + d] =
            sigmoidf_(aog[v] + bog) * (aop[v] + bop);
    }
}

// ---------------------------------------------------------------------------
extern "C" void kernel_launch(void* const* d_in, const int* in_sizes, int n_in,
                              void* d_out, int out_size, void* d_ws, size_t ws_size,
                              hipStream_t stream)
{
    const float*         z    = (const float*)d_in[0];
    const unsigned char* mask = (const unsigned char*)d_in[1];
    const float* nrmw = (const float*)d_in[2];
    const float* nrmb = (const float*)d_in[3];
    const float* lpw  = (const float*)d_in[4];
    const float* lpb  = (const float*)d_in[5];
    const float* lgw  = (const float*)d_in[6];
    const float* lgb  = (const float*)d_in[7];
    const float* rpw  = (const float*)d_in[8];
    const float* rpb  = (const float*)d_in[9];
    const float* rgw  = (const float*)d_in[10];
    const float* rgb  = (const float*)d_in[11];
    const float* onw  = (const float*)d_in[12];
    const float* onb  = (const float*)d_in[13];
    const float* opw  = (const float*)d_in[14];
    const float* opb  = (const float*)d_in[15];
    const float* ogw  = (const float*)d_in[16];
    const float* ogb  = (const float*)d_in[17];

    float* out = (float*)d_out;

    const int B = in_sizes[0] / (kN * kN * kD);   // = 4
    const size_t sliceElems = (size_t)kN * kN;    // 65536
    float* Lw = (float*)d_ws;                     // [B,128,N,N]
    float* Rw = Lw + (size_t)B * kD * sliceElems; // [B,128,N,N]  (268 MB total)

    k_norm_proj<<<dim3(B * 4096), dim3(256), 0, stream>>>(
        z, mask, nrmw, nrmb, lpw, lpb, lgw, lgb, rpw, rpb, rgw, rgb, Lw, Rw);

    k_tri<<<dim3(4, 4, B * kD), dim3(256), 0, stream>>>(Lw, Rw, out);

    k_out<<<dim3(B * 4096), dim3(256), 0, stream>>>(
        z, onw, onb, opw, opb, ogw, ogb, out);
}